// DynamicGraphConstruction_17695265259979
// MI455X (gfx1250) — compile-verified
//
#include <hip/hip_runtime.h>
#include <hip/hip_bf16.h>

#define BB 4
#define NSN 4096
#define NDN 4096
#define DD 256
#define KK 16
#define EE (BB*NSN*KK)      // 262144 edges
#define EPSV 1e-5f
#define TS 260              // padded LDS row stride (floats) -> conflict-free

typedef float v2f __attribute__((ext_vector_type(2)));
typedef float v8f __attribute__((ext_vector_type(8)));

// ---------------------------------------------------------------- tnorm ----
__global__ void tnorm_kernel(const float* __restrict__ dst_x,
                             float* __restrict__ tnorm) {
    int wid  = (blockIdx.x * blockDim.x + threadIdx.x) >> 5;   // one wave / row
    int lane = threadIdx.x & 31;
    if (wid >= BB * NDN) return;
    const float* row = dst_x + (size_t)wid * DD;
    float s = 0.f;
#pragma unroll
    for (int i = 0; i < 8; ++i) {
        float v = row[lane + 32 * i];
        s += v * v;
    }
#pragma unroll
    for (int off = 16; off > 0; off >>= 1) s += __shfl_xor(s, off, 32);
    if (lane == 0) tnorm[wid] = s;
}

// ----------------------------------------------------------------- init ----
__global__ void init_kernel(float* __restrict__ scal) {
    if (threadIdx.x < 4) scal[threadIdx.x] = 0.0f;
}

// ---------------------------------------------- fused GEMM + top-K (WMMA) ---
__global__ __launch_bounds__(128)
void knn_kernel(const float* __restrict__ src_x,
                const float* __restrict__ dst_x,
                const float* __restrict__ tnorm,
                float* __restrict__ out_idx,   // d_out: [2*EE] edge ids (as float)
                float* __restrict__ wraw) {    // ws: raw dot per edge
    __shared__ float Tsh[2][16 * TS];          // double-buffered T tiles
    __shared__ float scr[4][16][17];           // per-wave 16x16 score scratch

    const int tid  = threadIdx.x;
    const int wave = tid >> 5;
    const int lane = tid & 31;
    const int m0   = blockIdx.x * 64 + wave * 16;   // global src row base
    const int b    = m0 / NSN;
    const int cb   = b * NDN;                       // global dst base for batch

    // ---- load A fragments once: 16 rows x 256 f32 -> 64 x v2f in VGPRs ----
    const int arow = m0 + (lane & 15);
    const int koff = (lane >> 4) * 2;
    const float* aptr = src_x + (size_t)arow * DD + koff;
    v2f afrag[64];
#pragma unroll
    for (int kk = 0; kk < 64; ++kk) {
        afrag[kk].x = aptr[4 * kk];
        afrag[kk].y = aptr[4 * kk + 1];
    }

    // ---- streaming top-16 state (lanes 0..15, row = m0+lane) ----
    float sc[KK]; int id[KK];
#pragma unroll
    for (int i = 0; i < KK; ++i) { sc[i] = -__builtin_inff(); id[i] = 0; }
    float curMin = -__builtin_inff(); int minPos = 0;

    // async-LDS staging mapping: thread covers (trow, 8 x float4 chunks)
    const int trow = tid >> 3, tseg = tid & 7;
    const unsigned ldsBase =
        (unsigned)(size_t)(&Tsh[0][0]) + (unsigned)(trow * TS * 4);
    const float* grow0 = dst_x + (size_t)(cb + trow) * DD;

    // stage tile `t` into buffer `sel` via async global->LDS DMA (ASYNCcnt)
#define STAGE(t, sel)                                                        \
    do {                                                                     \
        unsigned lb = ldsBase + (unsigned)((sel) * 16 * TS * 4);             \
        const float* gp = grow0 + (size_t)(t) * 16 * DD;                     \
        _Pragma("unroll")                                                    \
        for (int i = 0; i < 8; ++i) {                                        \
            int f4 = i * 8 + tseg;                                           \
            asm volatile("global_load_async_to_lds_b128 %0, %1, off"         \
                         :: "v"(lb + (unsigned)(f4 * 16)), "v"(gp + f4 * 4)  \
                         : "memory");                                        \
        }                                                                    \
    } while (0)

    STAGE(0, 0);                                  // prologue: tile 0 -> buf 0

    for (int t = 0; t < NDN / 16; ++t) {
        const int cur = t & 1;
        const int n0  = t * 16;
        if (t + 1 < NDN / 16) {
            STAGE(t + 1, 1 - cur);                // prefetch next tile
            asm volatile("s_wait_asynccnt 0x8" ::: "memory");  // tile t landed
        } else {
            asm volatile("s_wait_asynccnt 0x0" ::: "memory");
        }
        __syncthreads();                          // all waves' tile-t data in LDS

        // ---- 16x16x256 tile via 64x v_wmma_f32_16x16x4_f32 ----
        v8f acc = {};
        const float* bbase = &Tsh[cur][(lane & 15) * TS + koff];
#pragma unroll
        for (int kk = 0; kk < 64; ++kk) {
            v2f bf;
            bf.x = bbase[4 * kk];
            bf.y = bbase[4 * kk + 1];
            acc = __builtin_amdgcn_wmma_f32_16x16x4_f32(
                false, afrag[kk], false, bf, (short)0, acc, false, false);
        }

        // scores = 2*dot - ||t||^2  -> per-wave LDS scratch [M][N]
        float tn = tnorm[cb + n0 + (lane & 15)];
        int part = lane >> 4;
#pragma unroll
        for (int r = 0; r < 8; ++r)
            scr[wave][r + 8 * part][lane & 15] = 2.0f * acc[r] - tn;
        __syncthreads();                 // everyone done reading Tsh[cur] + scr ready

        // streaming top-16 update (one lane per src row)
        if (lane < 16) {
#pragma unroll
            for (int j = 0; j < 16; ++j) {
                float s = scr[wave][lane][j];
                if (s > curMin) {
#pragma unroll
                    for (int i = 0; i < KK; ++i)
                        if (i == minPos) { sc[i] = s; id[i] = n0 + j; }
                    curMin = __builtin_inff();
#pragma unroll
                    for (int i = 0; i < KK; ++i)
                        if (sc[i] < curMin) { curMin = sc[i]; minPos = i; }
                }
            }
        }
    }
#undef STAGE

    // ---- sort descending (nearest first, matches top_k) & emit ----
    if (lane < 16) {
        int grow = m0 + lane;
#pragma unroll
        for (int k = 0; k < KK; ++k) {
            float best = -__builtin_inff(); int bpos = 0;
#pragma unroll
            for (int i = 0; i < KK; ++i)
                if (sc[i] > best) { best = sc[i]; bpos = i; }
            int bid = 0;
#pragma unroll
            for (int i = 0; i < KK; ++i)
                if (i == bpos) { bid = id[i]; sc[i] = -__builtin_inff(); }
            int gdst  = cb + bid;
            float dot = 0.5f * (best + tnorm[gdst]);   // recover raw dot
            int e = grow * KK + k;
            out_idx[e]       = (float)grow;            // edge_index[0]
            out_idx[EE + e]  = (float)gdst;            // edge_index[1]
            wraw[e] = dot;
        }
    }
}

// ------------------------------------------------------------- reduction ---
__global__ void reduce_kernel(const float* __restrict__ wraw,
                              float* __restrict__ scal) {
    __shared__ float sh0[256], sh1[256];
    int tid = threadIdx.x;
    float s = 0.f, s2 = 0.f;
    for (int i = blockIdx.x * 256 + tid; i < EE; i += 256 * 256) {
        float v = wraw[i]; s += v; s2 += v * v;
    }
    sh0[tid] = s; sh1[tid] = s2;
    __syncthreads();
    for (int off = 128; off > 0; off >>= 1) {
        if (tid < off) { sh0[tid] += sh0[tid + off]; sh1[tid] += sh1[tid + off]; }
        __syncthreads();
    }
    if (tid == 0) { atomicAdd(&scal[0], sh0[0]); atomicAdd(&scal[1], sh1[0]); }
}

// ------------------------------------------------- batchnorm + exp + sum ---
__global__ void expnorm_kernel(const float* __restrict__ wraw,
                               const float* __restrict__ gamma,
                               const float* __restrict__ beta,
                               float* __restrict__ scal,
                               float* __restrict__ zbuf) {
    __shared__ float sh[256];
    float mu  = scal[0] * (1.0f / (float)EE);
    float var = scal[1] * (1.0f / (float)EE) - mu * mu;
    float inv = rsqrtf(var + EPSV);
    float g  = gamma[0] * inv;
    float be = beta[0] - mu * g;               // z = exp(dot*g + be)
    float s = 0.f;
    for (int i = blockIdx.x * blockDim.x + threadIdx.x; i < EE;
         i += gridDim.x * blockDim.x) {
        float z = expf(wraw[i] * g + be);
        zbuf[i] = z; s += z;
    }
    sh[threadIdx.x] = s; __syncthreads();
    for (int off = 128; off > 0; off >>= 1) {
        if (threadIdx.x < off) sh[threadIdx.x] += sh[threadIdx.x + off];
        __syncthreads();
    }
    if (threadIdx.x == 0) atomicAdd(&scal[2], sh[0]);
}

// ------------------------------------------------------ unit-mean scaling --
__global__ void scale_kernel(const float* __restrict__ zbuf,
                             const float* __restrict__ scal,
                             float* __restrict__ wout) {
    int i = blockIdx.x * blockDim.x + threadIdx.x;
    if (i < EE) wout[i] = zbuf[i] * ((float)EE / scal[2]);
}

// ------------------------------------------------------------------ launch -
extern "C" void kernel_launch(void* const* d_in, const int* in_sizes, int n_in,
                              void* d_out, int out_size, void* d_ws, size_t ws_size,
                              hipStream_t stream) {
    (void)in_sizes; (void)n_in; (void)out_size; (void)ws_size;
    const float* src_x = (const float*)d_in[0];
    const float* dst_x = (const float*)d_in[1];
    // d_in[2], d_in[3] (src_batch/dst_batch, int64) unused: uniform blocks
    const float* gamma = (const float*)d_in[4];
    const float* beta  = (const float*)d_in[5];
    float* out = (float*)d_out;

    float* ws    = (float*)d_ws;
    float* scal  = ws;                 // [4]: sum, sumsq, sumexp, pad
    float* tnorm = ws + 64;            // [B*ND]
    float* wraw  = tnorm + BB * NDN;   // [EE] raw dots
    float* zbuf  = wraw + EE;          // [EE] exp values

    init_kernel<<<1, 32, 0, stream>>>(scal);
    tnorm_kernel<<<(BB * NDN) / 8, 256, 0, stream>>>(dst_x, tnorm);
    knn_kernel<<<(BB * NSN) / 64, 128, 0, stream>>>(src_x, dst_x, tnorm,
                                                    out, wraw);
    reduce_kernel<<<256, 256, 0, stream>>>(wraw, scal);
    expnorm_kernel<<<256, 256, 0, stream>>>(wraw, gamma, beta, scal, zbuf);
    scale_kernel<<<(EE + 255) / 256, 256, 0, stream>>>(zbuf, scal, out + 2 * EE);
}